// NCC_64441689309643
// MI455X (gfx1250) — compile-verified
//
#include <hip/hip_runtime.h>
#include <hip/hip_bf16.h>
#include <math.h>

typedef __attribute__((ext_vector_type(2))) float v2f;
typedef __attribute__((ext_vector_type(4))) float v4f;
typedef __attribute__((ext_vector_type(8))) float v8f;

#define NCC_EPS 1e-5f
#define NCC_BATCH 4
#define NCC_N 7077888            /* 192^3 elements per batch */
#define NCC_N4 (NCC_N / 4)       /* float4 elements per batch */
#define BLOCKS_PER_BATCH 512
#define NCC_THREADS 256
#define NCC_WAVES (NCC_THREADS / 32)

// Exact-f32 full-wave (32 lane) sum using V_WMMA_F32_16X16X4_F32.
// A: VGPR0 = lane value, VGPR1 = 0  ->  A[m][0]=v_m, A[m][2]=v_{m+16}, others 0.
// B: all ones (layout-independent). D[m][n] = v_m + v_{m+16} for every n.
// C/D layout: lanes 0-15 hold rows 0-7 in c[0..7], lanes 16-31 hold rows 8-15.
// Sum c[0..7] then fold halves with shfl_xor(16) -> full wave sum in all lanes.
__device__ __forceinline__ float wave_reduce_wmma(float v) {
    v2f a; a[0] = v;    a[1] = 0.0f;
    v2f b; b[0] = 1.0f; b[1] = 1.0f;
    v8f c = {};
    c = __builtin_amdgcn_wmma_f32_16x16x4_f32(
        /*neg_a=*/false, a, /*neg_b=*/false, b,
        /*c_mod=*/(short)0, c, /*reuse_a=*/false, /*reuse_b=*/false);
    float s = ((c[0] + c[1]) + (c[2] + c[3])) + ((c[4] + c[5]) + (c[6] + c[7]));
    s += __shfl_xor(s, 16, 32);
    return s;
}

// Pass 1: streaming partial sums. One group of blocks per batch; each block
// writes 5 deterministic partials (I, J, I*I, J*J, I*J) to d_ws.
__global__ __launch_bounds__(NCC_THREADS)
void ncc_partial_kernel(const float* __restrict__ y_pred,
                        const float* __restrict__ y_true,
                        float* __restrict__ partials) {
    const int batch = blockIdx.x / BLOCKS_PER_BATCH;
    const int blk   = blockIdx.x % BLOCKS_PER_BATCH;

    // Reference: Ii = y_true, Ji = y_pred
    const v4f* Ip = (const v4f*)(y_true + (size_t)batch * NCC_N);
    const v4f* Jp = (const v4f*)(y_pred + (size_t)batch * NCC_N);

    float sI = 0.f, sJ = 0.f, sII = 0.f, sJJ = 0.f, sIJ = 0.f;

    const int stride = BLOCKS_PER_BATCH * NCC_THREADS;
    for (int idx = blk * NCC_THREADS + (int)threadIdx.x; idx < NCC_N4; idx += stride) {
        // Streamed once: non-temporal b128 loads, don't thrash L2.
        v4f a = __builtin_nontemporal_load(&Ip[idx]);
        v4f b = __builtin_nontemporal_load(&Jp[idx]);
#pragma unroll
        for (int k = 0; k < 4; ++k) {
            float x = a[k], y = b[k];
            sI += x;
            sJ += y;
            sII = fmaf(x, x, sII);
            sJJ = fmaf(y, y, sJJ);
            sIJ = fmaf(x, y, sIJ);
        }
    }

    // Wave-level reduction on the matrix pipe (EXEC is all-1s here:
    // NCC_N4 % (32*stride-chunk) keeps whole waves convergent).
    sI  = wave_reduce_wmma(sI);
    sJ  = wave_reduce_wmma(sJ);
    sII = wave_reduce_wmma(sII);
    sJJ = wave_reduce_wmma(sJJ);
    sIJ = wave_reduce_wmma(sIJ);

    __shared__ float red[NCC_WAVES][5];
    const int wave = threadIdx.x >> 5;
    if ((threadIdx.x & 31) == 0) {
        red[wave][0] = sI;  red[wave][1] = sJ;  red[wave][2] = sII;
        red[wave][3] = sJJ; red[wave][4] = sIJ;
    }
    __syncthreads();
    if (threadIdx.x < 5) {
        float s = 0.f;
#pragma unroll
        for (int w = 0; w < NCC_WAVES; ++w) s += red[w][threadIdx.x];
        partials[((size_t)batch * BLOCKS_PER_BATCH + blk) * 5 + threadIdx.x] = s;
    }
}

// Pass 2: one block per batch folds the 512 block-partials in a fixed order
// and emits cc[batch]. Deterministic (no atomics anywhere).
__global__ __launch_bounds__(NCC_THREADS)
void ncc_final_kernel(const float* __restrict__ partials,
                      float* __restrict__ out) {
    const int batch = blockIdx.x;
    float acc[5] = {0.f, 0.f, 0.f, 0.f, 0.f};

    for (int i = threadIdx.x; i < BLOCKS_PER_BATCH; i += NCC_THREADS) {
        const float* p = partials + ((size_t)batch * BLOCKS_PER_BATCH + i) * 5;
#pragma unroll
        for (int j = 0; j < 5; ++j) acc[j] += p[j];
    }

#pragma unroll
    for (int j = 0; j < 5; ++j) acc[j] = wave_reduce_wmma(acc[j]);

    __shared__ float red[NCC_WAVES][5];
    const int wave = threadIdx.x >> 5;
    if ((threadIdx.x & 31) == 0) {
#pragma unroll
        for (int j = 0; j < 5; ++j) red[wave][j] = acc[j];
    }
    __syncthreads();

    if (threadIdx.x == 0) {
        float t[5];
#pragma unroll
        for (int j = 0; j < 5; ++j) {
            float s = 0.f;
#pragma unroll
            for (int w = 0; w < NCC_WAVES; ++w) s += red[w][j];
            t[j] = s;
        }
        const float W = (float)NCC_N;
        const float I_sum = t[0], J_sum = t[1];
        const float I2 = t[2], J2 = t[3], IJ = t[4];
        const float uI = I_sum / W, uJ = J_sum / W;
        const float cross = IJ - uJ * I_sum - uI * J_sum + uI * uJ * W;
        const float I_var = I2 - 2.f * uI * I_sum + uI * uI * W;
        const float J_var = J2 - 2.f * uJ * J_sum + uJ * uJ * W;
        out[batch] = cross / (sqrtf(I_var) * sqrtf(J_var) + NCC_EPS);
    }
}

extern "C" void kernel_launch(void* const* d_in, const int* in_sizes, int n_in,
                              void* d_out, int out_size, void* d_ws, size_t ws_size,
                              hipStream_t stream) {
    (void)in_sizes; (void)n_in; (void)out_size; (void)ws_size;
    const float* y_pred = (const float*)d_in[0];
    const float* y_true = (const float*)d_in[1];
    float* out = (float*)d_out;
    float* partials = (float*)d_ws;   // 4 * 512 * 5 floats = 40 KB scratch

    ncc_partial_kernel<<<NCC_BATCH * BLOCKS_PER_BATCH, NCC_THREADS, 0, stream>>>(
        y_pred, y_true, partials);
    ncc_final_kernel<<<NCC_BATCH, NCC_THREADS, 0, stream>>>(partials, out);
}